// HGNN_58128087384462
// MI455X (gfx1250) — compile-verified
//
#include <hip/hip_runtime.h>
#include <math.h>

#define N_NODES 100000
#define N_EDGES 50000
#define N_INC   1600000
#define N_GRAPHS 256
#define IN_DIM  128
#define HID     64
#define BN_EPS  1e-5f

typedef float v2f __attribute__((ext_vector_type(2)));
typedef float v8f __attribute__((ext_vector_type(8)));

// Only check for the WMMA builtin in the DEVICE pass; the host (x86) pass
// reports false for amdgcn aux-target builtins even though they are callable
// from device code.
#if defined(__HIP_DEVICE_COMPILE__) && defined(__AMDGCN__)
#if !__has_builtin(__builtin_amdgcn_wmma_f32_16x16x4_f32)
#error "missing __builtin_amdgcn_wmma_f32_16x16x4_f32 on device"
#endif
#endif

// ---------------------------------------------------------------------------
// WMMA GEMM: Y[M x 64] = X[M x K] @ W[K x 64]  (+ optional epilogues)
// EPI: 0 = none, 1 = relu(y + bias), 2 = relu(y + bias) + res
// One 256-thread block = 8 waves; W cached in LDS; each wave owns a 16-row
// M tile and a full 16x64 output strip (4 accumulators of v8f).
// Uses V_WMMA_F32_16X16X4_F32 (fp32 in / fp32 acc), K-loop step 4.
// ---------------------------------------------------------------------------
template<int K, int EPI>
__global__ __launch_bounds__(256) void gemm_wmma(
    const float* __restrict__ X, const float* __restrict__ W,
    const float* __restrict__ bias, const float* __restrict__ res,
    float* __restrict__ Y, int M)
{
    __shared__ float sW[K * HID];
    const int tid = threadIdx.x;
    for (int i = tid; i < K * HID; i += 256) sW[i] = W[i];
    __syncthreads();

    const int wave = tid >> 5;
    const int lane = tid & 31;
    const int tile = blockIdx.x * 8 + wave;
    const int Mtiles = M >> 4;
    if (tile >= Mtiles) return;

    const int nlo  = lane & 15;
    const int koff = (lane >> 4) << 1;          // lanes 0-15 -> K+0/1, 16-31 -> K+2/3
    const float* Arow = X + (size_t)(tile * 16 + nlo) * K;

    v8f acc[4] = {};
    for (int k = 0; k < K; k += 4) {
        v2f a;
        a.x = Arow[k + koff];
        a.y = Arow[k + koff + 1];
        const int kb = (k + koff) * HID;
        #pragma unroll
        for (int nt = 0; nt < 4; ++nt) {
            const int col = nt * 16 + nlo;
            v2f b;
            b.x = sW[kb + col];
            b.y = sW[kb + HID + col];
            acc[nt] = __builtin_amdgcn_wmma_f32_16x16x4_f32(
                false, a, false, b, (short)0, acc[nt], false, false);
        }
    }

    // C/D layout: VGPR r -> M=r (lanes 0-15) / M=r+8 (lanes 16-31), N=lane&15
    const int mbase = tile * 16 + ((lane >> 4) << 3);
    #pragma unroll
    for (int r = 0; r < 8; ++r) {
        const size_t row = (size_t)(mbase + r);
        float* yr = Y + row * HID;
        const float* rr = (EPI == 2) ? (res + row * HID) : nullptr;
        #pragma unroll
        for (int nt = 0; nt < 4; ++nt) {
            const int col = nt * 16 + nlo;
            float v = acc[nt][r];
            if (EPI >= 1) v = fmaxf(v + bias[col], 0.0f);
            if (EPI == 2) v += rr[col];
            yr[col] = v;
        }
    }
}

// ---------------------------------------------------------------------------
__global__ void fill_f32(float* p, float v, size_t n) {
    size_t i = (size_t)blockIdx.x * blockDim.x + threadIdx.x;
    if (i < n) p[i] = v;
}

__global__ void degree_kernel(const int* __restrict__ node, const int* __restrict__ edge,
                              float* deg, float* ecnt) {
    int i = blockIdx.x * blockDim.x + threadIdx.x;
    if (i < N_INC) {
        atomicAdd(&deg[node[i]], 1.0f);
        atomicAdd(&ecnt[edge[i]], 1.0f);
    }
}

// node -> hyperedge gather+scatter: thread = (incidence, feature)
__global__ void scatter_n2e(const int* __restrict__ node, const int* __restrict__ edge,
                            const float* __restrict__ xl, float* __restrict__ efeat) {
    size_t idx = (size_t)blockIdx.x * blockDim.x + threadIdx.x;
    if (idx >= (size_t)N_INC * HID) return;
    int i = (int)(idx >> 6), c = (int)(idx & 63);
    atomicAdd(&efeat[(size_t)edge[i] * HID + c], xl[(size_t)node[i] * HID + c]);
}

// hyperedge -> node
__global__ void scatter_e2n(const int* __restrict__ node, const int* __restrict__ edge,
                            const float* __restrict__ efeat, float* __restrict__ outb) {
    size_t idx = (size_t)blockIdx.x * blockDim.x + threadIdx.x;
    if (idx >= (size_t)N_INC * HID) return;
    int i = (int)(idx >> 6), c = (int)(idx & 63);
    atomicAdd(&outb[(size_t)node[i] * HID + c], efeat[(size_t)edge[i] * HID + c]);
}

// efeat[e][c] *= 1/ecnt[e]
__global__ void scale_edges(float* __restrict__ efeat, const float* __restrict__ ecnt) {
    size_t idx = (size_t)blockIdx.x * blockDim.x + threadIdx.x;
    if (idx >= (size_t)N_EDGES * HID) return;
    float c = ecnt[idx >> 6];
    efeat[idx] *= (c > 0.0f) ? (1.0f / c) : 0.0f;
}

// buf = buf * d_inv + bias; accumulate per-column sum / sumsq into stats[0..63]/[64..127]
__global__ void scale_bias_stats(float* __restrict__ buf, const float* __restrict__ deg,
                                 const float* __restrict__ bias, float* __restrict__ stats) {
    __shared__ float ssum[HID], ssq[HID];
    int t = threadIdx.x;
    if (t < HID) { ssum[t] = 0.0f; ssq[t] = 0.0f; }
    __syncthreads();
    size_t idx = (size_t)blockIdx.x * blockDim.x + t;
    if (idx < (size_t)N_NODES * HID) {
        int n = (int)(idx >> 6), c = (int)(idx & 63);
        float d = deg[n];
        float dinv = (d > 0.0f) ? (1.0f / d) : 0.0f;
        float v = buf[idx] * dinv + bias[c];
        buf[idx] = v;
        atomicAdd(&ssum[c], v);
        atomicAdd(&ssq[c], v * v);
    }
    __syncthreads();
    if (t < HID) {
        atomicAdd(&stats[t], ssum[t]);
        atomicAdd(&stats[HID + t], ssq[t]);
    }
}

// BatchNorm (biased batch stats) + ReLU in place
__global__ void bn_relu(float* __restrict__ buf, const float* __restrict__ stats,
                        const float* __restrict__ g, const float* __restrict__ b) {
    size_t idx = (size_t)blockIdx.x * blockDim.x + threadIdx.x;
    if (idx >= (size_t)N_NODES * HID) return;
    int c = (int)(idx & 63);
    const float invN = 1.0f / (float)N_NODES;
    float m = stats[c] * invN;
    float var = stats[HID + c] * invN - m * m;
    float v = (buf[idx] - m) * rsqrtf(var + BN_EPS) * g[c] + b[c];
    buf[idx] = fmaxf(v, 0.0f);
}

__device__ float atomicMaxF(float* addr, float val) {
    unsigned int* ua = (unsigned int*)addr;
    unsigned int old = *ua;
    while (true) {
        float f = __uint_as_float(old);
        if (f >= val) break;
        unsigned int assumed = old;
        old = atomicCAS(ua, assumed, __float_as_uint(val));
        if (old == assumed) break;
    }
    return __uint_as_float(old);
}

// gate[n] = h[n] . Wg + bg ; gmax[batch[n]] = max(...)
__global__ void gate_kernel(const float* __restrict__ h, const float* __restrict__ Wg,
                            const float* __restrict__ bg, const int* __restrict__ batch,
                            float* __restrict__ gate, float* __restrict__ gmax) {
    int n = blockIdx.x * blockDim.x + threadIdx.x;
    if (n >= N_NODES) return;
    const float* hr = h + (size_t)n * HID;
    float acc = bg[0];
    #pragma unroll
    for (int c = 0; c < HID; ++c) acc += hr[c] * Wg[c];
    gate[n] = acc;
    atomicMaxF(&gmax[batch[n]], acc);
}

__global__ void eg_kernel(const float* __restrict__ gate, const int* __restrict__ batch,
                          const float* __restrict__ gmax, float* __restrict__ eg,
                          float* __restrict__ denom) {
    int n = blockIdx.x * blockDim.x + threadIdx.x;
    if (n >= N_NODES) return;
    int g = batch[n];
    float e = expf(gate[n] - gmax[g]);
    eg[n] = e;
    atomicAdd(&denom[g], e);
}

__global__ void pool_kernel(const float* __restrict__ h, const float* __restrict__ eg,
                            const int* __restrict__ batch, float* __restrict__ pooled) {
    size_t idx = (size_t)blockIdx.x * blockDim.x + threadIdx.x;
    if (idx >= (size_t)N_NODES * HID) return;
    int n = (int)(idx >> 6), c = (int)(idx & 63);
    atomicAdd(&pooled[(size_t)batch[n] * HID + c], eg[n] * h[idx]);
}

// out[g][k] = (pooled[g]/denom[g]) . Wfc[:,k] + bfc[k]
__global__ void out_kernel(const float* __restrict__ pooled, const float* __restrict__ denom,
                           const float* __restrict__ Wfc, const float* __restrict__ bfc,
                           float* __restrict__ out) {
    int idx = blockIdx.x * blockDim.x + threadIdx.x;
    if (idx >= N_GRAPHS * 2) return;
    int g = idx >> 1, k = idx & 1;
    float acc = 0.0f;
    const float* pr = pooled + (size_t)g * HID;
    #pragma unroll
    for (int c = 0; c < HID; ++c) acc += pr[c] * Wfc[c * 2 + k];
    out[idx] = acc / denom[g] + bfc[k];
}

// ---------------------------------------------------------------------------
extern "C" void kernel_launch(void* const* d_in, const int* in_sizes, int n_in,
                              void* d_out, int out_size, void* d_ws, size_t ws_size,
                              hipStream_t stream) {
    const float* x    = (const float*)d_in[0];
    const int*  hidx  = (const int*)d_in[1];
    const int*  node  = hidx;
    const int*  edge  = hidx + N_INC;
    const int*  batch = (const int*)d_in[2];
    const float* W1   = (const float*)d_in[3];
    const float* b1   = (const float*)d_in[4];
    const float* bn1g = (const float*)d_in[5];
    const float* bn1b = (const float*)d_in[6];
    const float* W2   = (const float*)d_in[7];
    const float* b2   = (const float*)d_in[8];
    const float* bn2g = (const float*)d_in[9];
    const float* bn2b = (const float*)d_in[10];
    const float* Wh1  = (const float*)d_in[11];
    const float* bh1  = (const float*)d_in[12];
    const float* Wh2  = (const float*)d_in[13];
    const float* bh2  = (const float*)d_in[14];
    const float* Wg   = (const float*)d_in[15];
    const float* bg   = (const float*)d_in[16];
    const float* Wfc  = (const float*)d_in[17];
    const float* bfc  = (const float*)d_in[18];
    float* out = (float*)d_out;

    // workspace carve-up
    float* ws    = (float*)d_ws;
    float* bufA  = ws;                          // 6.4M  (xl / t1)
    float* bufB  = bufA + (size_t)N_NODES*HID;  // 6.4M  (h1, then hsum)
    float* bufC  = bufB + (size_t)N_NODES*HID;  // 6.4M  (h2)
    float* efeat = bufC + (size_t)N_NODES*HID;  // 3.2M
    float* deg   = efeat + (size_t)N_EDGES*HID; // 100k
    float* ecnt  = deg + N_NODES;               // 50k  (contiguous with deg)
    float* gate  = ecnt + N_EDGES;              // 100k
    float* eg    = gate + N_NODES;              // 100k
    float* stats = eg + N_NODES;                // 128
    float* gmax  = stats + 2*HID;               // 256
    float* denom = gmax + N_GRAPHS;             // 256
    float* pooled= denom + N_GRAPHS;            // 16384 (contiguous with denom)

    const int TB = 256;
    const size_t elemsN = (size_t)N_NODES * HID;   // 6,400,000
    const size_t elemsE = (size_t)N_EDGES * HID;   // 3,200,000
    const size_t elemsI = (size_t)N_INC * HID;     // 102,400,000
    const int gN  = (int)((elemsN + TB - 1) / TB); // 25000
    const int gE  = (int)((elemsE + TB - 1) / TB); // 12500
    const int gI  = (int)((elemsI + TB - 1) / TB); // 400000
    const int gInc = (N_INC + TB - 1) / TB;        // 6250
    const int gNode = (N_NODES + TB - 1) / TB;     // 391
    const int gGemm = ((N_NODES >> 4) + 7) / 8;    // 782 (6250 tiles / 8 waves)

    // degrees (shared by both conv layers)
    fill_f32<<<(N_NODES + N_EDGES + TB - 1) / TB, TB, 0, stream>>>(deg, 0.0f, N_NODES + N_EDGES);
    degree_kernel<<<gInc, TB, 0, stream>>>(node, edge, deg, ecnt);

    // ---- hconv 1: bufB = relu(bn(D^-1 H B^-1 H^T (x@W1) + b1)) ----
    gemm_wmma<IN_DIM, 0><<<gGemm, TB, 0, stream>>>(x, W1, nullptr, nullptr, bufA, N_NODES);
    fill_f32<<<gE, TB, 0, stream>>>(efeat, 0.0f, elemsE);
    scatter_n2e<<<gI, TB, 0, stream>>>(node, edge, bufA, efeat);
    scale_edges<<<gE, TB, 0, stream>>>(efeat, ecnt);
    fill_f32<<<gN, TB, 0, stream>>>(bufB, 0.0f, elemsN);
    fill_f32<<<1, TB, 0, stream>>>(stats, 0.0f, 2 * HID);
    scatter_e2n<<<gI, TB, 0, stream>>>(node, edge, efeat, bufB);
    scale_bias_stats<<<gN, TB, 0, stream>>>(bufB, deg, b1, stats);
    bn_relu<<<gN, TB, 0, stream>>>(bufB, stats, bn1g, bn1b);

    // ---- hconv 2: bufC = relu(bn(... (bufB@W2) ... + b2)) ----
    gemm_wmma<HID, 0><<<gGemm, TB, 0, stream>>>(bufB, W2, nullptr, nullptr, bufA, N_NODES);
    fill_f32<<<gE, TB, 0, stream>>>(efeat, 0.0f, elemsE);
    scatter_n2e<<<gI, TB, 0, stream>>>(node, edge, bufA, efeat);
    scale_edges<<<gE, TB, 0, stream>>>(efeat, ecnt);
    fill_f32<<<gN, TB, 0, stream>>>(bufC, 0.0f, elemsN);
    fill_f32<<<1, TB, 0, stream>>>(stats, 0.0f, 2 * HID);
    scatter_e2n<<<gI, TB, 0, stream>>>(node, edge, efeat, bufC);
    scale_bias_stats<<<gN, TB, 0, stream>>>(bufC, deg, b2, stats);
    bn_relu<<<gN, TB, 0, stream>>>(bufC, stats, bn2g, bn2b);

    // ---- head: bufA = relu(h@Wh1+bh1);  bufB = relu(bufA@Wh2+bh2) + h ----
    gemm_wmma<HID, 1><<<gGemm, TB, 0, stream>>>(bufC, Wh1, bh1, nullptr, bufA, N_NODES);
    gemm_wmma<HID, 2><<<gGemm, TB, 0, stream>>>(bufA, Wh2, bh2, bufC, bufB, N_NODES);

    // ---- attentional pooling over sorted batch ----
    fill_f32<<<1, TB, 0, stream>>>(gmax, -INFINITY, N_GRAPHS);
    fill_f32<<<(N_GRAPHS + N_GRAPHS * HID + TB - 1) / TB, TB, 0, stream>>>(
        denom, 0.0f, N_GRAPHS + (size_t)N_GRAPHS * HID);
    gate_kernel<<<gNode, TB, 0, stream>>>(bufB, Wg, bg, batch, gate, gmax);
    eg_kernel<<<gNode, TB, 0, stream>>>(gate, batch, gmax, eg, denom);
    pool_kernel<<<gN, TB, 0, stream>>>(bufB, eg, batch, pooled);
    out_kernel<<<(N_GRAPHS * 2 + TB - 1) / TB, TB, 0, stream>>>(pooled, denom, Wfc, bfc, out);
}